// ChamferDistance_16243566313485
// MI455X (gfx1250) — compile-verified
//
#include <hip/hip_runtime.h>

typedef __attribute__((ext_vector_type(2))) float v2f;
typedef __attribute__((ext_vector_type(8))) float v8f;

// Nearest-neighbor argmin via V_WMMA_F32_16X16X4_F32.
// For each query point x_i (rows), find argmin_j of ||x_i - y_j||^2.
// We minimize score = ||y_j||^2 - 2*x_i.y_j  (||x_i||^2 is a per-row constant:
// dropping it cannot change the argmin or tie structure).
//
// Wave-level mapping (wave32):
//   A operand (16x4 f32, 2 VGPRs): lane L<16 -> M=L, (K0,K1)=(x.c0,x.c1)
//                                  lane L>=16 -> M=L-16, (K2,K3)=(x.c2, 0)
//   B operand (4x16 f32, 2 VGPRs): identical per-lane packing with N=L%16.
//   C/D (8 VGPRs): vgpr v, lanes 0-15 -> (M=v,   N=lane)
//                          lanes 16-31 -> (M=v+8, N=lane-16)
//
// Depth-2 software pipeline: the global_load_b96 issued at iteration t is
// consumed at iteration t+2, so s_wait_loadcnt sinks ~2 iterations below the
// load (tile index clamped with wave-uniform scalar math -> no divergence,
// EXEC stays all-1s at every WMMA as the ISA requires).
__global__ __launch_bounds__(256) void chamfer_nn_wmma(
    const float* __restrict__ X,   // [B, nq, 3] queries
    const float* __restrict__ Y,   // [B, nr, 3] references
    int* __restrict__ out,         // [B, nq]
    int nq, int nr, int totalWaves)
{
  const int lane      = threadIdx.x & 31;
  const int waveInBlk = threadIdx.x >> 5;
  const int w         = blockIdx.x * 8 + waveInBlk;
  if (w >= totalWaves) return;                 // wave-uniform: EXEC stays all-1s

  const int qtPerB  = nq >> 4;
  const int b       = w / qtPerB;
  const int qt      = w - b * qtPerB;
  const int q0      = qt << 4;
  const int halfSel = lane >> 4;               // 0: lanes 0-15, 1: lanes 16-31
  const int lm      = lane & 15;

  // ---- Load A operand once: this wave's 16 query points ----
  const float* xp = X + ((size_t)b * nq + (q0 + lm)) * 3;
  const float xc0 = xp[0], xc1 = xp[1], xc2 = xp[2];
  v2f A;
  A.x = halfSel ? xc2 : xc0;
  A.y = halfSel ? 0.0f : xc1;

  float bestv[8];
  int   besti[8];
#pragma unroll
  for (int v = 0; v < 8; ++v) { bestv[v] = 3.0e38f; besti[v] = 0; }

  const float* Yb     = Y + (size_t)b * nr * 3;
  const int    rtiles = nr >> 4;

  // ---- Prologue: preload tiles 0 and 1 (clamped for tiny nr) ----
  float cur0, cur1, cur2, nx0, nx1, nx2;
  {
    const float* p0 = Yb + (size_t)lm * 3;
    cur0 = p0[0]; cur1 = p0[1]; cur2 = p0[2];
    const int t1 = (1 < rtiles) ? 1 : 0;       // uniform clamp
    const float* p1 = Yb + (size_t)(t1 * 16 + lm) * 3;
    nx0 = p1[0]; nx1 = p1[1]; nx2 = p1[2];
  }

#pragma unroll 2
  for (int t = 0; t < rtiles; ++t) {
    // ---- Issue load for tile t+2 (consumed two iterations from now) ----
    int tn = t + 2;
    if (tn >= rtiles) tn = rtiles - 1;         // uniform clamp, stays in bounds
    const float* pn = Yb + (size_t)(tn * 16 + lm) * 3;
    __builtin_prefetch(pn + 48 * 6, 0, 1);     // speculative, ~6 tiles further
    const float f0 = pn[0], f1 = pn[1], f2 = pn[2];

    // ---- Compute with tile t (data loaded two iterations ago) ----
    const float sq = cur0 * cur0 + cur1 * cur1 + cur2 * cur2;  // ||y_N||^2
    v2f Bv;
    Bv.x = halfSel ? cur2 : cur0;
    Bv.y = halfSel ? 0.0f : cur1;

    v8f C = {0.f, 0.f, 0.f, 0.f, 0.f, 0.f, 0.f, 0.f};
    // D = A(16x4) * B(4x16) + 0 : 16x16 tile of cross = x_M . y_N
    C = __builtin_amdgcn_wmma_f32_16x16x4_f32(
        /*neg_a=*/false, A, /*neg_b=*/false, Bv,
        /*c_mod=*/(short)0, C, /*reuse_a=*/false, /*reuse_b=*/false);

    const int cand = t * 16 + lm;
#pragma unroll
    for (int v = 0; v < 8; ++v) {
      const float score  = __builtin_fmaf(-2.0f, C[v], sq);
      const bool  better = score < bestv[v];   // strict: earliest index wins
      bestv[v] = better ? score : bestv[v];
      besti[v] = better ? cand  : besti[v];
    }

    // ---- Rotate pipeline registers (renamed away under unroll) ----
    cur0 = nx0; cur1 = nx1; cur2 = nx2;
    nx0 = f0;  nx1 = f1;  nx2 = f2;
  }

  // ---- Argmin reduction over the 16 lanes of each half (xor butterfly) ----
#pragma unroll
  for (int v = 0; v < 8; ++v) {
    float val = bestv[v];
    int   idx = besti[v];
#pragma unroll
    for (int m = 1; m < 16; m <<= 1) {
      const float ov = __shfl_xor(val, m, 16);
      const int   oi = __shfl_xor(idx, m, 16);
      if (ov < val || (ov == val && oi < idx)) { val = ov; idx = oi; }
    }
    bestv[v] = val;
    besti[v] = idx;
  }

  // Lane 0 writes rows M=0..7, lane 16 writes rows M=8..15.
  if (lm == 0) {
    int* op = out + (size_t)b * nq + (q0 + halfSel * 8);
#pragma unroll
    for (int v = 0; v < 8; ++v) op[v] = besti[v];
  }
}

extern "C" void kernel_launch(void* const* d_in, const int* in_sizes, int n_in,
                              void* d_out, int out_size, void* d_ws, size_t ws_size,
                              hipStream_t stream) {
  (void)n_in; (void)d_ws; (void)ws_size; (void)out_size;
  const float* xyz1 = (const float*)d_in[0];
  const float* xyz2 = (const float*)d_in[1];
  int* out = (int*)d_out;   // int32 indices (4-byte elements)

  const int B = 4;
  const int n = in_sizes[0] / (B * 3);   // 8192
  const int m = in_sizes[1] / (B * 3);   // 8192

  // Direction 1: idx1[b,i] = argmin_j ||xyz1[b,i] - xyz2[b,j]||^2
  {
    const int waves  = B * (n >> 4);
    const int blocks = (waves + 7) / 8;
    chamfer_nn_wmma<<<blocks, 256, 0, stream>>>(xyz1, xyz2, out, n, m, waves);
  }
  // Direction 2: idx2[b,j] = argmin_i ||xyz2[b,j] - xyz1[b,i]||^2
  {
    const int waves  = B * (m >> 4);
    const int blocks = (waves + 7) / 8;
    chamfer_nn_wmma<<<blocks, 256, 0, stream>>>(xyz2, xyz1, out + (size_t)B * n, m, n, waves);
  }
}